// PostProcessor_33878702031444
// MI455X (gfx1250) — compile-verified
//
#include <hip/hip_runtime.h>
#include <math.h>

// ---------------------------------------------------------------------------
// Faster-RCNN PostProcessor for MI455X (gfx1250, wave32).
// No matmul structure exists in this workload, so no WMMA: the CDNA5-specific
// paths used are the async-to-LDS DMA engine (ASYNCcnt) for staging + gathering
// NMS data, s_wait_asynccnt, global_prefetch_b8, and wave32-sized workgroups.
// ---------------------------------------------------------------------------

#define NPROP 1000
#define NPAD  1024      // padded per-class row count (power of two for bitonic)
#define NCLS  81
#define NFG   80
#define CAND_PER_CLASS 128
#define NCAND (NFG * CAND_PER_CLASS)
#define DETS  100
#define IMG_W 1333.0f
#define IMG_H 800.0f
#define SCORE_TH 0.05f
#define NMS_TH   0.5f
#define XFORM_CLIP 4.135166556742356f   // log(1000/16)

__device__ __forceinline__ float neg_inf() { return __int_as_float(0xff800000); }

// ---- CDNA5 async-to-LDS helpers (ASYNCcnt engine) --------------------------
// Per-lane addressed: each lane copies global memory at its vaddr to LDS at its vdst -> hardware gather into
// LDS with no VGPR data movement. Tracked by ASYNCcnt (s_wait_asynccnt).
__device__ __forceinline__ void async_ld_b32(unsigned lds_off, const void* g) {
  unsigned long long ga = (unsigned long long)g;
  asm volatile("global_load_async_to_lds_b32 %0, %1, off"
               :: "v"(lds_off), "v"(ga) : "memory");
}
__device__ __forceinline__ void async_ld_b128(unsigned lds_off, const void* g) {
  unsigned long long ga = (unsigned long long)g;
  asm volatile("global_load_async_to_lds_b128 %0, %1, off"
               :: "v"(lds_off), "v"(ga) : "memory");
}
__device__ __forceinline__ void wait_async0() {
  asm volatile("s_wait_asynccnt 0x0" ::: "memory");
}

// ---------------------------------------------------------------------------
// Kernel 1: softmax + box decode + clip, scatter to class-major buffers.
// score_buf[cf*1024 + r] : prob if > thresh else -inf  (rows 1000..1023 = -inf)
// box_buf [(cf*1024 + r)*4 + k] : clipped xyxy
// ---------------------------------------------------------------------------
__global__ void __launch_bounds__(256) decode_score_kernel(
    const float* __restrict__ logits, const float* __restrict__ regr,
    const float* __restrict__ props, float* __restrict__ score_buf,
    float* __restrict__ box_buf) {
  int r = blockIdx.x * blockDim.x + threadIdx.x;
  if (r >= NPAD) return;
  if (r >= NPROP) {  // padding rows
    for (int cf = 0; cf < NFG; ++cf) {
      score_buf[cf * NPAD + r] = neg_inf();
      float* b = box_buf + (size_t)(cf * NPAD + r) * 4;
      b[0] = b[1] = b[2] = b[3] = 0.f;
    }
    return;
  }
  const float* lrow = logits + r * NCLS;
  const float* drow = regr + r * 4 * NCLS;
  __builtin_prefetch(lrow, 0, 1);         // -> global_prefetch_b8
  __builtin_prefetch(drow, 0, 1);

  // softmax over 81 classes
  float m = lrow[0];
  for (int c = 1; c < NCLS; ++c) m = fmaxf(m, lrow[c]);
  float s = 0.f;
  for (int c = 0; c < NCLS; ++c) s += __expf(lrow[c] - m);
  float inv = 1.f / s;

  float px1 = props[r * 4 + 0], py1 = props[r * 4 + 1];
  float px2 = props[r * 4 + 2], py2 = props[r * 4 + 3];
  float w = px2 - px1 + 1.f, h = py2 - py1 + 1.f;
  float cx = px1 + 0.5f * w, cy = py1 + 0.5f * h;

  for (int c = 1; c < NCLS; ++c) {
    int cf = c - 1;
    float prob = __expf(lrow[c] - m) * inv;
    score_buf[cf * NPAD + r] = (prob > SCORE_TH) ? prob : neg_inf();

    float dx = drow[4 * c + 0] * 0.1f;
    float dy = drow[4 * c + 1] * 0.1f;
    float dw = fminf(drow[4 * c + 2] * 0.2f, XFORM_CLIP);
    float dh = fminf(drow[4 * c + 3] * 0.2f, XFORM_CLIP);
    float pcx = dx * w + cx, pcy = dy * h + cy;
    float pw = __expf(dw) * w, ph = __expf(dh) * h;
    float x1 = pcx - 0.5f * pw, y1 = pcy - 0.5f * ph;
    float x2 = pcx + 0.5f * pw - 1.f, y2 = pcy + 0.5f * ph - 1.f;
    x1 = fminf(fmaxf(x1, 0.f), IMG_W - 1.f);
    y1 = fminf(fmaxf(y1, 0.f), IMG_H - 1.f);
    x2 = fminf(fmaxf(x2, 0.f), IMG_W - 1.f);
    y2 = fminf(fmaxf(y2, 0.f), IMG_H - 1.f);
    float* b = box_buf + (size_t)(cf * NPAD + r) * 4;
    b[0] = x1; b[1] = y1; b[2] = x2; b[3] = y2;
  }
}

// ---------------------------------------------------------------------------
// Kernel 2: per-class sort + greedy NMS + compaction into top-128 candidates.
// One 1024-thread (32-wave) block per foreground class. 32 KB static LDS.
// Single __shared__ object per kernel => it sits at LDS offset 0, so
// offsetof() gives the LDS byte address needed by the async-to-LDS ops.
// ---------------------------------------------------------------------------
struct SortSmem {
  alignas(16) float box[NPAD][4];  // 16 KB (16B-aligned for b128 async writes)
  float score[NPAD];               // 4 KB
  int   idx[NPAD];                 // 4 KB
  int   keep[NPAD];                // 4 KB
  int   pref[NPAD];                // 4 KB
};

__global__ void __launch_bounds__(1024) nms_kernel(
    const float* __restrict__ score_buf, const float* __restrict__ box_buf,
    float* __restrict__ cand_score, float* __restrict__ cand_box) {
  __shared__ SortSmem sm;
  const int tid = threadIdx.x;
  const int cf  = blockIdx.x;

  // --- async DMA stage: scores -> LDS (bulk, contiguous) ---
  unsigned score_off = (unsigned)__builtin_offsetof(SortSmem, score) + (unsigned)tid * 4u;
  async_ld_b32(score_off, score_buf + (size_t)cf * NPAD + tid);
  sm.idx[tid] = tid;
  wait_async0();
  __syncthreads();

  // --- bitonic sort: descending score, stable tie-break on original index ---
  for (int k = 2; k <= NPAD; k <<= 1) {
    for (int j = k >> 1; j > 0; j >>= 1) {
      __syncthreads();
      int p = tid ^ j;
      if (p > tid) {
        float a = sm.score[tid], b = sm.score[p];
        int ia = sm.idx[tid], ib = sm.idx[p];
        bool before = (a > b) || (a == b && ia < ib);  // "a comes first"
        bool up = ((tid & k) == 0);
        if (before != up) {
          sm.score[tid] = b; sm.score[p] = a;
          sm.idx[tid]   = ib; sm.idx[p]  = ia;
        }
      }
    }
  }
  __syncthreads();

  // --- async DMA gather: boxes by sorted index -> LDS (per-lane addressed) ---
  unsigned box_off = (unsigned)__builtin_offsetof(SortSmem, box) + (unsigned)tid * 16u;
  async_ld_b128(box_off, box_buf + (size_t)(cf * NPAD + sm.idx[tid]) * 4);
  wait_async0();

  int fin = (sm.score[tid] > neg_inf()) ? 1 : 0;
  sm.keep[tid] = fin;
  int n_valid = __syncthreads_count(fin);   // barrier: async LDS writes visible

  float bx1 = sm.box[tid][0], by1 = sm.box[tid][1];
  float bx2 = sm.box[tid][2], by2 = sm.box[tid][3];
  float area = (bx2 - bx1 + 1.f) * (by2 - by1 + 1.f);

  // --- greedy suppression, exactly the reference fori_loop semantics ---
  for (int i = 0; i < n_valid; ++i) {
    if (sm.keep[i] && tid > i && sm.keep[tid]) {
      float ix1 = sm.box[i][0], iy1 = sm.box[i][1];
      float ix2 = sm.box[i][2], iy2 = sm.box[i][3];
      float xx1 = fmaxf(ix1, bx1), yy1 = fmaxf(iy1, by1);
      float xx2 = fminf(ix2, bx2), yy2 = fminf(iy2, by2);
      float iw = fmaxf(xx2 - xx1 + 1.f, 0.f);
      float ih = fmaxf(yy2 - yy1 + 1.f, 0.f);
      float inter = iw * ih;
      float areai = (ix2 - ix1 + 1.f) * (iy2 - iy1 + 1.f);
      float uni = fmaxf(area + areai - inter, 1e-6f);
      if (inter > NMS_TH * uni) sm.keep[tid] = 0;
    }
    __syncthreads();
  }

  // --- compact kept entries (sorted by score) into per-class candidates ---
  sm.pref[tid] = sm.keep[tid];
  __syncthreads();
  for (int off = 1; off < NPAD; off <<= 1) {   // Hillis-Steele inclusive scan
    int v = (tid >= off) ? sm.pref[tid - off] : 0;
    __syncthreads();
    sm.pref[tid] += v;
    __syncthreads();
  }
  int rank = sm.pref[tid] - sm.keep[tid];

  if (tid < CAND_PER_CLASS) {
    cand_score[cf * CAND_PER_CLASS + tid] = neg_inf();
    float* cb = cand_box + (size_t)(cf * CAND_PER_CLASS + tid) * 4;
    cb[0] = cb[1] = cb[2] = cb[3] = 0.f;
  }
  __threadfence_block();
  __syncthreads();
  if (sm.keep[tid] && rank < CAND_PER_CLASS) {
    int q = cf * CAND_PER_CLASS + rank;
    cand_score[q] = sm.score[tid];
    float* cb = cand_box + (size_t)q * 4;
    cb[0] = bx1; cb[1] = by1; cb[2] = bx2; cb[3] = by2;
  }
}

// ---------------------------------------------------------------------------
// Kernel 3: global top-100 over 80*128 candidates (each class already sorted,
// so >100 candidates per class can never reach the global top-100).
// 100 rounds of parallel argmax; picked entries marked NaN so they can't repeat.
// Output: dets[100][5] = {x1,y1,x2,y2,score}, then labels[100] (as float).
// ---------------------------------------------------------------------------
struct TopkSmem {
  float sc[NCAND];    // 40 KB
  float rv[1024];     // 4 KB
  int   ri[1024];     // 4 KB
};

__global__ void __launch_bounds__(1024) topk_kernel(
    const float* __restrict__ cand_score, const float* __restrict__ cand_box,
    float* __restrict__ out) {
  __shared__ TopkSmem sm;
  const int tid = threadIdx.x;
  for (int i = tid; i < NCAND; i += 1024) sm.sc[i] = cand_score[i];
  __syncthreads();

  for (int k = 0; k < DETS; ++k) {
    float bv = 0.f; int bi = -1;
    for (int i = tid; i < NCAND; i += 1024) {
      float v = sm.sc[i];
      bool ok = (v == v) && (bi < 0 || v > bv || (v == bv && i < bi));
      if (ok) { bv = v; bi = i; }
    }
    sm.rv[tid] = bv; sm.ri[tid] = bi;
    __syncthreads();
    for (int s = 512; s > 0; s >>= 1) {
      if (tid < s) {
        float v = sm.rv[tid + s]; int ix = sm.ri[tid + s];
        bool ok = (ix >= 0) &&
                  (sm.ri[tid] < 0 || v > sm.rv[tid] ||
                   (v == sm.rv[tid] && ix < sm.ri[tid]));
        if (ok) { sm.rv[tid] = v; sm.ri[tid] = ix; }
      }
      __syncthreads();
    }
    if (tid == 0) {
      int b = sm.ri[0];
      float v = (b >= 0) ? sm.rv[0] : neg_inf();
      float x1 = 0.f, y1 = 0.f, x2 = 0.f, y2 = 0.f;
      int label = 0;
      if (b >= 0) {
        x1 = cand_box[b * 4 + 0]; y1 = cand_box[b * 4 + 1];
        x2 = cand_box[b * 4 + 2]; y2 = cand_box[b * 4 + 3];
        if (v > neg_inf()) label = b / CAND_PER_CLASS + 1;
        sm.sc[b] = __int_as_float(0x7fc00000);  // NaN: never selected again
      }
      out[k * 5 + 0] = x1; out[k * 5 + 1] = y1;
      out[k * 5 + 2] = x2; out[k * 5 + 3] = y2;
      out[k * 5 + 4] = v;
      out[5 * DETS + k] = (float)label;
    }
    __syncthreads();
  }
}

// ---------------------------------------------------------------------------
extern "C" void kernel_launch(void* const* d_in, const int* in_sizes, int n_in,
                              void* d_out, int out_size, void* d_ws, size_t ws_size,
                              hipStream_t stream) {
  (void)in_sizes; (void)n_in; (void)out_size; (void)ws_size;
  const float* logits = (const float*)d_in[0];   // [1000, 81]
  const float* regr   = (const float*)d_in[1];   // [1000, 324]
  const float* props  = (const float*)d_in[2];   // [1000, 4]

  float* ws = (float*)d_ws;
  float* score_buf  = ws;                              // 80*1024
  float* box_buf    = ws + (size_t)NFG * NPAD;         // 80*1024*4
  float* cand_score = ws + (size_t)NFG * NPAD * 5;     // 80*128
  float* cand_box   = cand_score + NCAND;              // 80*128*4
  float* out = (float*)d_out;                          // 600 floats

  decode_score_kernel<<<4, 256, 0, stream>>>(logits, regr, props, score_buf, box_buf);
  nms_kernel<<<NFG, NPAD, 0, stream>>>(score_buf, box_buf, cand_score, cand_box);
  topk_kernel<<<1, 1024, 0, stream>>>(cand_score, cand_box, out);
}